// AttentionLstm_2714419331403
// MI455X (gfx1250) — compile-verified
//
#include <hip/hip_runtime.h>
#include <hip/hip_bf16.h>

// Problem constants from the reference
#define BB 128
#define SS 512
#define DD 512
// H=4, DH=128 -- irrelevant: softmax over a size-1 axis == 1.0, so attn == v.

typedef __attribute__((ext_vector_type(2))) float v2f;
typedef __attribute__((ext_vector_type(8))) float v8f;

// ---------------------------------------------------------------------------
// Wave-level f32 WMMA GEMM strip: one wave computes a 16(M) x 64(N) strip of
// C = A(16x512) * B(512x64-slice) using V_WMMA_F32_16X16X4_F32, K-step = 4.
// A lane layout (ISA 7.12.2, 32-bit A 16x4): lane L holds row m=L&15,
// K pair kb = 2*(L>>4). B symmetric: col n=L&15, rows kb, kb+1.
// A fragment is reused across the 4 N-tiles (4 accumulators, 32 VGPRs).
// All strides are compile-time (512) so B loads become base + immediate.
// ---------------------------------------------------------------------------
__device__ __forceinline__ void wmma_strip4_f32(const float* __restrict__ Arow, // &A[row0+m][kb]
                                                const float* __restrict__ Bbase, // &B[kb][col0+m]
                                                v8f acc[4])
{
#pragma unroll 4
    for (int k0 = 0; k0 < DD; k0 += 4) {
        v2f a;
        a.x = Arow[k0];
        a.y = Arow[k0 + 1];
        const float* bp = Bbase + (long)k0 * DD;
#pragma unroll
        for (int j = 0; j < 4; ++j) {
            v2f b;
            b.x = bp[j * 16];        // row kb   , col col0 + j*16 + m
            b.y = bp[DD + j * 16];   // row kb+1 , same col
            acc[j] = __builtin_amdgcn_wmma_f32_16x16x4_f32(
                /*neg_a=*/false, a, /*neg_b=*/false, b,
                /*c_mod=*/(short)0, acc[j], /*reuse_a=*/false, /*reuse_b=*/false);
        }
    }
}

// ---------------------------------------------------------------------------
// C[M x 512] = A[M x 512] @ B[512 x 512] (+ bias if non-null).
// blockDim = 128 (4 waves). Block covers a 64(M) x 64(N) tile:
// wave w -> M-tile (blockIdx.y*4 + w), N-strip blockIdx.x (64 cols).
// Used for xf = x@Wi + bi  and  Wvo = Wv@Wo.
// ---------------------------------------------------------------------------
__global__ void gemm_bias_kernel(const float* __restrict__ A,
                                 const float* __restrict__ Bm,
                                 const float* __restrict__ bias,
                                 float* __restrict__ C)
{
    const unsigned lane = threadIdx.x & 31u;
    const unsigned wave = threadIdx.x >> 5u;
    const int row0 = ((int)blockIdx.y * 4 + (int)wave) * 16;
    const int col0 = (int)blockIdx.x * 64;
    const int m  = (int)(lane & 15u);
    const int kb = (int)((lane >> 4u) << 1u); // 0 or 2

    v8f acc[4] = {v8f{}, v8f{}, v8f{}, v8f{}};
    wmma_strip4_f32(A + (long)(row0 + m) * DD + kb,
                    Bm + (long)kb * DD + col0 + m, acc);

    const int n     = (int)(lane & 15u);
    const int rbase = row0 + (int)((lane >> 4u) << 3u); // +0 or +8
#pragma unroll
    for (int j = 0; j < 4; ++j) {
        const float bv = bias ? bias[col0 + j * 16 + n] : 0.0f;
#pragma unroll
        for (int i = 0; i < 8; ++i)
            C[(long)(rbase + i) * DD + col0 + j * 16 + n] = acc[j][i] + bv;
    }
}

// ---------------------------------------------------------------------------
// One recurrence step:  out[:,t,:] = h_prev @ Wvo + bvo + out[:,t,:]
// (residual term xf[:,t,:] already sits in out[:,t,:]; read-before-write by
// the owning thread). h_prev row stride = astride (DD for packed h0, SS*DD
// otherwise). M = BB = 128 -> grid = (8, 2), block = 128 (4 waves).
// ---------------------------------------------------------------------------
__global__ void step_kernel(const float* __restrict__ hprev, long astride,
                            const float* __restrict__ Wvo,
                            const float* __restrict__ bvo,
                            float* __restrict__ outt /* = out + t*DD */)
{
    const unsigned lane = threadIdx.x & 31u;
    const unsigned wave = threadIdx.x >> 5u;
    const int row0 = ((int)blockIdx.y * 4 + (int)wave) * 16; // batch-tile base
    const int col0 = (int)blockIdx.x * 64;
    const int m  = (int)(lane & 15u);
    const int kb = (int)((lane >> 4u) << 1u);

    v8f acc[4] = {v8f{}, v8f{}, v8f{}, v8f{}};
    wmma_strip4_f32(hprev + (long)(row0 + m) * astride + kb,
                    Wvo + (long)kb * DD + col0 + m, acc);

    const int n     = (int)(lane & 15u);
    const int rbase = row0 + (int)((lane >> 4u) << 3u);
#pragma unroll
    for (int j = 0; j < 4; ++j) {
        const float bv = bvo[col0 + j * 16 + n];
#pragma unroll
        for (int i = 0; i < 8; ++i) {
            const long idx = (long)(rbase + i) * (long)(SS * DD) + (col0 + j * 16 + n);
            outt[idx] = acc[j][i] + bv + outt[idx]; // read x_t, write h_{t+1}
        }
    }
}

// bvo[n] = bo[n] + sum_k bv[k] * Wo[k][n]
__global__ void bvo_kernel(const float* __restrict__ bvv,
                           const float* __restrict__ Wo,
                           const float* __restrict__ bo,
                           float* __restrict__ bvo)
{
    const int n = (int)blockIdx.x * (int)blockDim.x + (int)threadIdx.x;
    if (n < DD) {
        float s = bo[n];
        for (int k = 0; k < DD; ++k) s += bvv[k] * Wo[(long)k * DD + n];
        bvo[n] = s;
    }
}

// Pack h0 = xf[:,0,:] into a contiguous (BB x DD) buffer (the final-state
// region of d_out, reused as scratch; overwritten by the gather at the end).
__global__ void copy_h0_kernel(const float* __restrict__ out, float* __restrict__ h0)
{
    const int i = (int)blockIdx.x * (int)blockDim.x + (int)threadIdx.x; // 0..BB*DD-1
    const int b = i / DD, d = i % DD;
    h0[i] = out[(long)b * SS * DD + d];
}

// final_state[b,:] = outputs[b, data_lens[b]-1, :]
__global__ void final_gather_kernel(const float* __restrict__ out,
                                    const int* __restrict__ lens,
                                    float* __restrict__ fin)
{
    const int i = (int)blockIdx.x * (int)blockDim.x + (int)threadIdx.x; // 0..BB*DD-1
    const int b = i / DD, d = i % DD;
    const int t = lens[b] - 1;
    fin[i] = out[(long)b * SS * DD + (long)t * DD + d];
}

extern "C" void kernel_launch(void* const* d_in, const int* in_sizes, int n_in,
                              void* d_out, int out_size, void* d_ws, size_t ws_size,
                              hipStream_t stream)
{
    (void)in_sizes; (void)n_in; (void)out_size; (void)ws_size;

    const float* x   = (const float*)d_in[0];
    const float* Wi  = (const float*)d_in[1];
    const float* bi  = (const float*)d_in[2];
    // d_in[3..6] = Wq, bq, Wk, bk  -- mathematically dead (softmax over size-1 axis == 1)
    const float* Wv  = (const float*)d_in[7];
    const float* bvv = (const float*)d_in[8];
    const float* Wo  = (const float*)d_in[9];
    const float* bo  = (const float*)d_in[10];
    const int*  lens = (const int*)d_in[11];

    float* out  = (float*)d_out;                      // outputs: BB*SS*DD floats
    float* fin  = out + (long)BB * SS * DD;           // final_state: BB*DD floats
    float* Wvo  = (float*)d_ws;                       // 512*512 floats
    float* bvo  = Wvo + (long)DD * DD;                // 512 floats

    dim3 blk(128, 1, 1);

    // 1) xf = x @ Wi + bi -> written straight into the outputs region of d_out
    //    M = BB*SS = 65536 rows -> grid = (512/64, 65536/64) = (8, 1024)
    gemm_bias_kernel<<<dim3(DD / 64, (BB * SS) / 64), blk, 0, stream>>>(x, Wi, bi, out);

    // 2) Wvo = Wv @ Wo   (M = 512 -> grid = (8, 8))
    gemm_bias_kernel<<<dim3(DD / 64, DD / 64), blk, 0, stream>>>(Wv, Wo, nullptr, Wvo);

    // 3) bvo = bv @ Wo + bo
    bvo_kernel<<<dim3(2), dim3(256), 0, stream>>>(bvv, Wo, bo, bvo);

    // 4) h0 = xf[:,0,:] packed into the final-state region (avoids the t=0 RAW race)
    copy_h0_kernel<<<dim3((BB * DD) / 256), dim3(256), 0, stream>>>(out, fin);

    // 5) sequential scan: out[:,t,:] = h_{t-1} @ Wvo + bvo + xf[:,t,:]
    for (int t = 0; t < SS; ++t) {
        const float* hprev  = (t == 0) ? (const float*)fin : (const float*)(out + (long)(t - 1) * DD);
        const long  astride = (t == 0) ? (long)DD : (long)SS * DD;
        step_kernel<<<dim3(DD / 64, BB / 64), blk, 0, stream>>>(
            hprev, astride, Wvo, bvo, out + (long)t * DD);
    }

    // 6) final_state gather (overwrites the h0 scratch with the real answer)
    final_gather_kernel<<<dim3((BB * DD) / 256), dim3(256), 0, stream>>>(out, lens, fin);
}